// DynamicGCN_69853348102243
// MI455X (gfx1250) — compile-verified
//
#include <hip/hip_runtime.h>

typedef __attribute__((ext_vector_type(2))) float v2f;
typedef __attribute__((ext_vector_type(8))) float v8f;

#define NN 50000
#define NE 640000

// ---------------------------------------------------------------------------
// Degree / normalization
// ---------------------------------------------------------------------------
__global__ __launch_bounds__(256) void deg_zero(float* deg) {
    int i = blockIdx.x * blockDim.x + threadIdx.x;
    if (i < NN) deg[i] = 0.0f;
}

__global__ __launch_bounds__(256) void deg_count(const int* __restrict__ ei, float* deg) {
    int e = blockIdx.x * blockDim.x + threadIdx.x;
    if (e < NE) atomicAdd(&deg[ei[NE + e]], 1.0f);  // dst row
}

// deg includes +1 self-loop; dis = rsqrt(deg) (always >= 1)
__global__ __launch_bounds__(256) void make_dis(float* deg) {
    int i = blockIdx.x * blockDim.x + threadIdx.x;
    if (i < NN) deg[i] = rsqrtf(deg[i] + 1.0f);
}

// ---------------------------------------------------------------------------
// WMMA fp32 GEMM:  C[M x NOUT] = A[M x 128] * W[128 x NOUT]
// One wave computes one 16x16 output tile, K swept 4 at a time with
// V_WMMA_F32_16X16X4_F32 (true fp32 -> matches reference precision).
// M = 50000 = 3125*16 exactly; NOUT in {128,64}; no remainders, EXEC all-1s.
// ---------------------------------------------------------------------------
template <int NOUT>
__global__ __launch_bounds__(256) void gemm_wmma_f32(const float* __restrict__ A,
                                                     const float* __restrict__ W,
                                                     float* __restrict__ C, int M) {
    constexpr int K = 128;
    constexpr int TN = NOUT / 16;
    const int wave = blockIdx.x * (blockDim.x >> 5) + (threadIdx.x >> 5);
    const int total = (M / 16) * TN;
    if (wave >= total) return;                 // uniform per wave -> EXEC stays all-1s

    const int tm = wave / TN;
    const int tn = wave % TN;
    const int lane = threadIdx.x & 31;
    const int half = lane >> 4;                // 0: K pair {k,k+1}; 1: {k+2,k+3}
    const int r    = lane & 15;                // A row / B column within tile

    const float* arow = A + (size_t)(tm * 16 + r) * K + 2 * half;
    const float* wcol = W + (size_t)(2 * half) * NOUT + tn * 16 + r;

    v8f acc = {};
#pragma unroll
    for (int k = 0; k < K; k += 4) {
        v2f a = *(const v2f*)(arow + k);       // A frag: {K=k(+2h), K=k+1(+2h)}
        v2f b;
        b.x = wcol[(size_t)k * NOUT];          // B frag: W[k+2h][n], W[k+2h+1][n]
        b.y = wcol[(size_t)k * NOUT + NOUT];
        acc = __builtin_amdgcn_wmma_f32_16x16x4_f32(
            /*neg_a=*/false, a, /*neg_b=*/false, b,
            /*c_mod=*/(short)0, acc, /*reuse_a=*/false, /*reuse_b=*/false);
    }

    // C/D layout: VGPR i -> M = i + 8*half, N = lane&15
    float* crow = C + (size_t)(tm * 16 + half * 8) * NOUT + tn * 16 + r;
#pragma unroll
    for (int i = 0; i < 8; ++i) crow[(size_t)i * NOUT] = acc[i];
}

// ---------------------------------------------------------------------------
// out[i*F + j] = bias[j]   (aggregation buffers start at bias)
// ---------------------------------------------------------------------------
template <int F>
__global__ __launch_bounds__(256) void init_bias(float* __restrict__ out,
                                                 const float* __restrict__ b) {
    int i = blockIdx.x * blockDim.x + threadIdx.x;
    if (i < NN * F) out[i] = b[i & (F - 1)];
}

// ---------------------------------------------------------------------------
// Edge scatter: one wave per edge (incl. self-loops appended at the end).
// lane handles F/32 contiguous floats: vector load from h[src], atomic adds
// into out[dst], scaled by norm = dis[src]*dis[dst].
// ---------------------------------------------------------------------------
template <int F>
__global__ __launch_bounds__(256) void scatter_edges(const float* __restrict__ h,
                                                     const int* __restrict__ ei,
                                                     const float* __restrict__ dis,
                                                     float* __restrict__ out) {
    constexpr int VEC = F / 32;
    const int gid  = blockIdx.x * blockDim.x + threadIdx.x;
    const int edge = gid >> 5;
    const int lane = gid & 31;
    if (edge >= NE + NN) return;

    int s, d;
    if (edge < NE) { s = ei[edge]; d = ei[NE + edge]; }
    else           { s = d = edge - NE; }

    const float nrm = dis[s] * dis[d];
    const float* hs = h + (size_t)s * F + lane * VEC;
    float*       od = out + (size_t)d * F + lane * VEC;

    if constexpr (VEC == 4) {
        float4 v = *(const float4*)hs;         // global_load_b128
        atomicAdd(od + 0, nrm * v.x);
        atomicAdd(od + 1, nrm * v.y);
        atomicAdd(od + 2, nrm * v.z);
        atomicAdd(od + 3, nrm * v.w);
    } else {
        float2 v = *(const float2*)hs;
        atomicAdd(od + 0, nrm * v.x);
        atomicAdd(od + 1, nrm * v.y);
    }
}

__global__ __launch_bounds__(256) void relu_k(float* __restrict__ x, int n) {
    int i = blockIdx.x * blockDim.x + threadIdx.x;
    if (i < n) x[i] = fmaxf(x[i], 0.0f);
}

// ---------------------------------------------------------------------------
// Launch
// ---------------------------------------------------------------------------
extern "C" void kernel_launch(void* const* d_in, const int* in_sizes, int n_in,
                              void* d_out, int out_size, void* d_ws, size_t ws_size,
                              hipStream_t stream) {
    const float* x  = (const float*)d_in[0];   // [50000,128]
    const int*   ei = (const int*)d_in[1];     // [2,640000] (src row, dst row)
    const float* W1 = (const float*)d_in[2];
    const float* b1 = (const float*)d_in[3];
    const float* W2 = (const float*)d_in[4];
    const float* b2 = (const float*)d_in[5];
    const float* W3 = (const float*)d_in[6];
    const float* b3 = (const float*)d_in[7];
    float* out = (float*)d_out;                // [50000,64]

    float* dis  = (float*)d_ws;                // NN floats (deg -> dis in place)
    float* bufA = dis + NN;                    // NN*128 floats
    float* bufB = bufA + (size_t)NN * 128;     // NN*128 floats

    const int ThB = 256;
    const int gN   = (NN + ThB - 1) / ThB;
    const int gE   = (NE + ThB - 1) / ThB;
    const int gNF  = (NN * 128 + ThB - 1) / ThB;
    const int gSc  = ((NE + NN) * 32 + ThB - 1) / ThB;
    const int gGm128 = ((NN / 16) * 8 + 7) / 8;   // 8 waves/block
    const int gGm64  = ((NN / 16) * 4 + 7) / 8;

    // --- normalization coefficients ---
    deg_zero<<<gN, ThB, 0, stream>>>(dis);
    deg_count<<<gE, ThB, 0, stream>>>(ei, dis);
    make_dis<<<gN, ThB, 0, stream>>>(dis);

    // --- layer 1: h1 = x@W1 (A); agg1 = scatter(h1)+b1, relu (B) ---
    gemm_wmma_f32<128><<<gGm128, ThB, 0, stream>>>(x, W1, bufA, NN);
    init_bias<128><<<gNF, ThB, 0, stream>>>(bufB, b1);
    scatter_edges<128><<<gSc, ThB, 0, stream>>>(bufA, ei, dis, bufB);
    relu_k<<<gNF, ThB, 0, stream>>>(bufB, NN * 128);

    // --- layer 2: h2 = agg1@W2 (A); agg2 = scatter(h2)+b2, relu (B) ---
    gemm_wmma_f32<128><<<gGm128, ThB, 0, stream>>>(bufB, W2, bufA, NN);
    init_bias<128><<<gNF, ThB, 0, stream>>>(bufB, b2);
    scatter_edges<128><<<gSc, ThB, 0, stream>>>(bufA, ei, dis, bufB);
    relu_k<<<gNF, ThB, 0, stream>>>(bufB, NN * 128);

    // --- layer 3: h3 = agg2@W3 (A, N x 64); out = scatter(h3)+b3 ---
    gemm_wmma_f32<64><<<gGm64, ThB, 0, stream>>>(bufB, W3, bufA, NN);
    init_bias<64><<<(NN * 64 + ThB - 1) / ThB, ThB, 0, stream>>>(out, b3);
    scatter_edges<64><<<gSc, ThB, 0, stream>>>(bufA, ei, dis, out);
}